// Deep_AD_58213986730480
// MI455X (gfx1250) — compile-verified
//
#include <hip/hip_runtime.h>
#include <hip/hip_bf16.h>

typedef float v2f __attribute__((ext_vector_type(2)));
typedef float v8f __attribute__((ext_vector_type(8)));

#define TW 32
#define TH 32
#define LSTRIDE 34
#define LDS_ROWS 36               // 34 real rows + 2 pad rows so tap-9..11 reads stay in-bounds/finite
#define LDS_ELEMS (LDS_ROWS * LSTRIDE)
#define HW 512
#define NIMG 16

// One diffusion step: out = in - (1/8) * sum_c g(d_c)*d_c,  d = conv3x3(in, W_t) + b_t
// Conv mapped onto V_WMMA_F32_16X16X4_F32:
//   A (16x4) = weights, M=channel (rows 0-7 real, 8-15 zero), K=taps (9 real, 10-11 zero)
//   B (4x16) = im2col patches, K=taps, N=16 pixels
//   D (16x16): lane n (n<16), VGPRs 0..7 hold channels 0..7 of pixel n -> in-lane channel sum
// B values for taps >= 9 are don't-cares (zero A weights) but must be FINITE (WMMA propagates
// NaN), hence the fully-staged padded LDS tile and unconditional tap loads.
__global__ __launch_bounds__(256) void ad_step_kernel(
    const float* __restrict__ in, const float* __restrict__ Wt,
    const float* __restrict__ bt, float* __restrict__ out)
{
    __shared__ float tile[LDS_ELEMS];

    const int tx  = blockIdx.x * TW;
    const int ty  = blockIdx.y * TH;
    const int img = blockIdx.z;
    const float* img_in  = in  + (size_t)img * HW * HW;
    float*       img_out = out + (size_t)img * HW * HW;

    // Stage 34x34 tile (1-px halo) + 2 pad rows into LDS; everything finite.
    for (int idx = threadIdx.x; idx < LDS_ELEMS; idx += 256) {
        int ly = idx / LSTRIDE, lx = idx % LSTRIDE;
        int gy = ty + ly - 1, gx = tx + lx - 1;
        float v = 0.0f;
        if (gy >= 0 && gy < HW && gx >= 0 && gx < HW)
            v = img_in[gy * HW + gx];
        tile[idx] = v;
    }
    __syncthreads();

    const int lane = threadIdx.x & 31;
    const int wave = threadIdx.x >> 5;
    const int hi   = lane >> 4;   // lane-half selects K pair (A/B 16x16x4 layout)
    const int lo   = lane & 15;   // M (channel) for A, N (pixel) for B/D

    // A fragments: 8 channels x 9 taps, padded to 16 x 12. Invariant per launch.
    v2f afrag[3];
    #pragma unroll
    for (int f = 0; f < 3; ++f) {
        int kb = f * 4 + hi * 2;
        afrag[f].x = (lo < 8 && kb     < 9) ? Wt[lo * 9 + kb]     : 0.0f;
        afrag[f].y = (lo < 8 && kb + 1 < 9) ? Wt[lo * 9 + kb + 1] : 0.0f;
    }
    // Bias folded into initial accumulator (uniform across lanes per acc row).
    float bias[8];
    #pragma unroll
    for (int r = 0; r < 8; ++r) bias[r] = bt[r];

    // Per-lane tap offsets, tap t -> (t/3)*LSTRIDE + t%3, taps {0,1,4,5,8,9} / {2,3,6,7,10,11}.
    const int toff0 = hi ? 2   : 0;
    const int toff1 = hi ? 34  : 1;
    const int toff2 = hi ? 68  : 35;
    const int toff3 = hi ? 69  : 36;
    const int toff4 = hi ? 103 : 70;
    const int toff5 = hi ? 104 : 102;

    // 8 waves cover 32 rows; each wave: rows {w, w+8, w+16, w+24} x 2 groups of 16 px.
    for (int row = wave; row < TH; row += 8) {
        #pragma unroll
        for (int g = 0; g < 2; ++g) {
            const int px   = g * 16 + lo;
            const int base = row * LSTRIDE + px;

            // Gather all six B taps up front (unconditional ds_loads).
            v2f b0, b1, b2;
            b0.x = tile[base + toff0];
            b0.y = tile[base + toff1];
            b1.x = tile[base + toff2];
            b1.y = tile[base + toff3];
            b2.x = tile[base + toff4];
            b2.y = tile[base + toff5];

            v8f acc;
            #pragma unroll
            for (int r = 0; r < 8; ++r) acc[r] = bias[r];

            acc = __builtin_amdgcn_wmma_f32_16x16x4_f32(
                    false, afrag[0], false, b0, (short)0, acc, false, false);
            acc = __builtin_amdgcn_wmma_f32_16x16x4_f32(
                    false, afrag[1], false, b1, (short)0, acc, false, false);
            acc = __builtin_amdgcn_wmma_f32_16x16x4_f32(
                    false, afrag[2], false, b2, (short)0, acc, false, false);

            // Epilogue: lanes 0-15 hold channels 0-7 of their pixel in acc[0..7].
            if (hi == 0) {
                float s = 0.0f;
                #pragma unroll
                for (int r = 0; r < 8; ++r) {
                    float d  = acc[r];
                    float rc = __builtin_amdgcn_rcpf(1.0f + d * d);   // v_rcp_f32, no IEEE div chain
                    s += d * __expf(-__builtin_fabsf(d) * rc);        // v_exp_f32 path
                }
                float center = tile[base + LSTRIDE + 1];
                img_out[(size_t)(ty + row) * HW + (tx + px)] = center - s * 0.125f;
            }
        }
    }
}

extern "C" void kernel_launch(void* const* d_in, const int* in_sizes, int n_in,
                              void* d_out, int out_size, void* d_ws, size_t ws_size,
                              hipStream_t stream) {
    const float* x = (const float*)d_in[0];   // [16,1,512,512]
    const float* W = (const float*)d_in[1];   // [3,8,1,3,3] -> 72 floats per step
    const float* b = (const float*)d_in[2];   // [3,8]
    float* out = (float*)d_out;
    float* ws  = (float*)d_ws;                // one 16.8MB intermediate image

    dim3 grid(HW / TW, HW / TH, NIMG);
    dim3 block(256);

    // Steps are globally dependent (conv reads updated neighbors) -> 3 launches,
    // ping-pong x -> out -> ws -> out.
    ad_step_kernel<<<grid, block, 0, stream>>>(x,   W + 0 * 72, b + 0 * 8, out);
    ad_step_kernel<<<grid, block, 0, stream>>>(out, W + 1 * 72, b + 1 * 8, ws);
    ad_step_kernel<<<grid, block, 0, stream>>>(ws,  W + 2 * 72, b + 2 * 8, out);
}